// RootNet_61744449847595
// MI455X (gfx1250) — compile-verified
//
#include <hip/hip_runtime.h>
#include <math.h>

// Problem constants (fixed by the reference)
#define NG   32768            // num_graphs
#define JPG  24               // joints per graph
#define NN   (NG * JPG)       // 786432 nodes
#define NE   1507328          // edges
#define LCAP 131072           // compacted-edge list capacity (expected ~62.8K survivors)

typedef __attribute__((ext_vector_type(2))) float v2f;
typedef __attribute__((ext_vector_type(8))) float v8f;

__device__ __forceinline__ float sigmoidf_(float x) {
    return 1.0f / (1.0f + expf(-x));
}
// jax.nn.softplus == logaddexp(x, 0) == max(x,0) + log1p(exp(-|x|))
__device__ __forceinline__ float softplusf_(float x) {
    return fmaxf(x, 0.0f) + log1pf(expf(-fabsf(x)));
}

// Hardware global_atomic_add_f32 (avoid CAS loop)
__device__ __forceinline__ void atomic_fadd(float* p, float v) {
    unsafeAtomicAdd(p, v);
}

// ---------------------------------------------------------------------------
// Zero the two compaction counters (called first every launch).
// ---------------------------------------------------------------------------
__global__ void k_zero(unsigned* __restrict__ c) {
    c[threadIdx.x] = 0u;   // 32 threads zero c[0..31]; counters at [0] and [16]
}

// ---------------------------------------------------------------------------
// Filter pass: compact edge IDs with dst % 24 == 0 into a dense list.
// Block-level LDS aggregation -> one global counter atomic per block.
// ---------------------------------------------------------------------------
__global__ void k_filter(const int* __restrict__ dst_row,
                         unsigned* __restrict__ cnt,
                         unsigned* __restrict__ list)
{
    __shared__ unsigned lcnt;
    __shared__ unsigned lbase;
    int tid = threadIdx.x;
    if (tid == 0) lcnt = 0u;
    __syncthreads();

    int e = blockIdx.x * blockDim.x + tid;
    bool pred = false;
    if (e < NE) pred = (dst_row[e] % JPG) == 0;

    unsigned loff = 0u;
    if (pred) loff = atomicAdd(&lcnt, 1u);          // LDS atomic
    __syncthreads();
    if (tid == 0) lbase = lcnt ? atomicAdd(cnt, lcnt) : 0u;
    __syncthreads();

    if (pred) {
        unsigned slot = lbase + loff;
        if (slot < LCAP) list[slot] = (unsigned)e;  // guard (unreachable in practice)
    }
}

// ---------------------------------------------------------------------------
// Kernel: r1[b,:16] = bu1 + x3[root(b)] @ Wu1   (dense, f32 WMMA 16x16x4)
// One wave = 16 graphs (M) x 16 channels (N), K = 3 padded to 4. Branchless
// operand setup -> EXEC all-1s at the WMMA.
// A layout (ISA 16x4 f32): lanes 0-15 VGPR0=K0, VGPR1=K1; lanes 16-31 K2,K3.
// B layout (4x16, mirrors A K-split): VGPR0 = rows K0|K2, VGPR1 = rows K1|K3.
// C/D layout: VGPR r -> (M = r + (lane>=16 ? 8 : 0), N = lane & 15).
// ---------------------------------------------------------------------------
__global__ void k_up1(const float* __restrict__ x,
                      const float* __restrict__ Wu1,
                      const float* __restrict__ bu1,
                      float* __restrict__ r1)
{
    int lane  = threadIdx.x & 31;
    int wave  = blockIdx.x * (blockDim.x >> 5) + (threadIdx.x >> 5); // 0..2047
    int g0    = wave << 4;
    int m     = lane & 15;
    bool hi   = lane >= 16;
    int graph = g0 + m;

    const float* xr = x + (size_t)graph * (JPG * 9) + 6;  // x[root, 6:9]
    float x0 = xr[hi ? 2 : 0];
    float x1 = xr[1];
    float w0 = Wu1[(hi ? 2 : 0) * 16 + m];
    float w1 = Wu1[16 + m];

    v2f a, b;
    a.x = x0;  a.y = hi ? 0.0f : x1;
    b.x = w0;  b.y = hi ? 0.0f : w1;

    float bias = bu1[m];
    v8f c;
#pragma unroll
    for (int r = 0; r < 8; ++r) c[r] = bias;

    v8f dacc = __builtin_amdgcn_wmma_f32_16x16x4_f32(false, a, false, b,
                                                     (short)0, c, false, false);

    int Mbase = g0 + (hi ? 8 : 0);
#pragma unroll
    for (int r = 0; r < 8; ++r)
        r1[(size_t)(Mbase + r) * 16 + m] = dacc[r];
}

// ---------------------------------------------------------------------------
// Dense block-1 messages over the compacted list: full lane occupancy.
// ---------------------------------------------------------------------------
__global__ void k_msg1c(const float* __restrict__ x,
                        const int*   __restrict__ ei,
                        const float* __restrict__ ea,
                        const unsigned* __restrict__ cnt,
                        const unsigned* __restrict__ list,
                        const float* __restrict__ Wf1, const float* __restrict__ bf1,
                        const float* __restrict__ Ws1, const float* __restrict__ bs1,
                        float* __restrict__ r1)
{
    unsigned n = cnt[0];
    if (n > LCAP) n = LCAP;
    unsigned i = blockIdx.x * blockDim.x + threadIdx.x;
    if (i >= n) return;

    int e = (int)list[i];
    int d = ei[NE + e];                 // dst = edge_index[1]
    int b = d / JPG;
    int s = ei[e];                      // src = edge_index[0]

    float z[9];
    const float* xi = x + (size_t)d * 9 + 6;
    const float* xj = x + (size_t)s * 9 + 6;
    z[0] = xi[0]; z[1] = xi[1]; z[2] = xi[2];
    z[3] = xj[0]; z[4] = xj[1]; z[5] = xj[2];
    const float* ep = ea + (size_t)e * 3;
    z[6] = ep[0]; z[7] = ep[1]; z[8] = ep[2];

#pragma unroll
    for (int c = 0; c < 16; ++c) {
        float f  = bf1[c];
        float sv = bs1[c];
#pragma unroll
        for (int k = 0; k < 9; ++k) {
            f  += z[k] * Wf1[k * 16 + c];   // uniform weight reads -> s-loads
            sv += z[k] * Ws1[k * 16 + c];
        }
        atomic_fadd(&r1[(size_t)b * 16 + c], sigmoidf_(f) * softplusf_(sv));
    }
}

// ---------------------------------------------------------------------------
// Kernel: out[b,:3] = bu2 + h_root[b] @ Wu2, h_root = [r1[b], tna[root(b)]]
// K = 19 padded to 20 -> 5 chained f32 WMMAs per wave (16 graphs).
// ---------------------------------------------------------------------------
__global__ void k_up2(const float* __restrict__ r1,
                      const float* __restrict__ tna,
                      const float* __restrict__ Wu2,
                      const float* __restrict__ bu2,
                      float* __restrict__ out)
{
    int lane  = threadIdx.x & 31;
    int wave  = blockIdx.x * (blockDim.x >> 5) + (threadIdx.x >> 5);
    int g0    = wave << 4;
    int m     = lane & 15;
    bool hi   = lane >= 16;
    int graph = g0 + m;

    float h[20];
#pragma unroll
    for (int k = 0; k < 16; ++k) h[k] = r1[(size_t)graph * 16 + k];
    const float* tr = tna + (size_t)graph * JPG * 3;   // t_node_attr at root
    h[16] = tr[0]; h[17] = tr[1]; h[18] = tr[2]; h[19] = 0.0f;

    float bias = (m < 3) ? bu2[m] : 0.0f;
    v8f c;
#pragma unroll
    for (int r = 0; r < 8; ++r) c[r] = bias;

#pragma unroll
    for (int kk = 0; kk < 20; kk += 4) {
        int k0 = hi ? (kk + 2) : kk;
        v2f a, b;
        a.x = h[k0];
        a.y = h[k0 + 1];
        // Wu2 is 19x3; branchless clamped-index loads, select to zero-pad
        bool v0 = (k0     < 19) && (m < 3);
        bool v1 = (k0 + 1 < 19) && (m < 3);
        float w0 = Wu2[v0 ? (k0 * 3 + m) : 0];
        float w1 = Wu2[v1 ? ((k0 + 1) * 3 + m) : 0];
        b.x = v0 ? w0 : 0.0f;
        b.y = v1 ? w1 : 0.0f;
        c = __builtin_amdgcn_wmma_f32_16x16x4_f32(false, a, false, b,
                                                  (short)0, c, false, false);
    }

    if (m < 3) {
        int Mbase = g0 + (hi ? 8 : 0);
#pragma unroll
        for (int r = 0; r < 8; ++r)
            out[(size_t)(Mbase + r) * 3 + m] = c[r];
    }
}

// ---------------------------------------------------------------------------
// Dense block-2 messages over the compacted list.
// h[n] built on the fly from r1[n/24] and t_node_attr[n].
// ---------------------------------------------------------------------------
__global__ void k_msg2c(const float* __restrict__ r1,
                        const float* __restrict__ tna,
                        const int*   __restrict__ ei,
                        const float* __restrict__ ea,
                        const unsigned* __restrict__ cnt,
                        const unsigned* __restrict__ list,
                        const float* __restrict__ Wf2, const float* __restrict__ bf2,
                        const float* __restrict__ Ws2, const float* __restrict__ bs2,
                        float* __restrict__ out)
{
    unsigned n = cnt[0];
    if (n > LCAP) n = LCAP;
    unsigned i = blockIdx.x * blockDim.x + threadIdx.x;
    if (i >= n) return;

    int e  = (int)list[i];
    int d  = ei[NE + e];
    int b  = d / JPG;
    int s  = ei[e];
    int sb = s / JPG;

    float z[41];
#pragma unroll
    for (int k = 0; k < 16; ++k) z[k] = r1[(size_t)b * 16 + k];
    { const float* t = tna + (size_t)d * 3; z[16] = t[0]; z[17] = t[1]; z[18] = t[2]; }
#pragma unroll
    for (int k = 0; k < 16; ++k) z[19 + k] = r1[(size_t)sb * 16 + k];
    { const float* t = tna + (size_t)s * 3; z[35] = t[0]; z[36] = t[1]; z[37] = t[2]; }
    { const float* t = ea  + (size_t)e * 3; z[38] = t[0]; z[39] = t[1]; z[40] = t[2]; }

#pragma unroll
    for (int c = 0; c < 3; ++c) {
        float f  = bf2[c];
        float sv = bs2[c];
#pragma unroll
        for (int k = 0; k < 41; ++k) {
            f  += z[k] * Wf2[k * 3 + c];
            sv += z[k] * Ws2[k * 3 + c];
        }
        atomic_fadd(&out[(size_t)b * 3 + c], sigmoidf_(f) * softplusf_(sv));
    }
}

// ---------------------------------------------------------------------------
extern "C" void kernel_launch(void* const* d_in, const int* in_sizes, int n_in,
                              void* d_out, int out_size, void* d_ws, size_t ws_size,
                              hipStream_t stream)
{
    const float* x   = (const float*)d_in[0];
    const int*   ei1 = (const int*)  d_in[1];
    const float* ea1 = (const float*)d_in[2];
    const float* tna = (const float*)d_in[3];
    const int*   ei2 = (const int*)  d_in[4];
    const float* ea2 = (const float*)d_in[5];
    const float* Wf1 = (const float*)d_in[6];
    const float* bf1 = (const float*)d_in[7];
    const float* Ws1 = (const float*)d_in[8];
    const float* bs1 = (const float*)d_in[9];
    const float* Wu1 = (const float*)d_in[10];
    const float* bu1 = (const float*)d_in[11];
    const float* Wf2 = (const float*)d_in[12];
    const float* bf2 = (const float*)d_in[13];
    const float* Ws2 = (const float*)d_in[14];
    const float* bs2 = (const float*)d_in[15];
    const float* Wu2 = (const float*)d_in[16];
    const float* bu2 = (const float*)d_in[17];

    float* out = (float*)d_out;

    // Workspace layout: r1 (2 MB) | counters (128 B) | list1 (512 KB) | list2 (512 KB)
    unsigned char* ws   = (unsigned char*)d_ws;
    float*    r1    = (float*)ws;
    unsigned* cnts  = (unsigned*)(ws + (size_t)NG * 16 * sizeof(float));
    unsigned* list1 = (unsigned*)(ws + (size_t)NG * 16 * sizeof(float) + 128);
    unsigned* list2 = list1 + LCAP;

    const int upBlocks  = (NG / 16) / 8;       // 2048 waves, 8 waves/block
    const int fBlocks   = (NE + 255) / 256;    // filter scan
    const int mBlocks   = LCAP / 256;          // dense message pass bound

    k_zero  <<<1, 32, 0, stream>>>(cnts);
    k_filter<<<fBlocks, 256, 0, stream>>>(ei1 + NE, &cnts[0],  list1);
    k_filter<<<fBlocks, 256, 0, stream>>>(ei2 + NE, &cnts[16], list2);
    k_up1   <<<upBlocks, 256, 0, stream>>>(x, Wu1, bu1, r1);
    k_msg1c <<<mBlocks, 256, 0, stream>>>(x, ei1, ea1, &cnts[0], list1,
                                          Wf1, bf1, Ws1, bs1, r1);
    k_up2   <<<upBlocks, 256, 0, stream>>>(r1, tna, Wu2, bu2, out);
    k_msg2c <<<mBlocks, 256, 0, stream>>>(r1, tna, ei2, ea2, &cnts[16], list2,
                                          Wf2, bf2, Ws2, bs2, out);
}